// NFLinearMlpHNet_86354612453619
// MI455X (gfx1250) — compile-verified
//
#include <hip/hip_runtime.h>
#include <math.h>

// Problem constants (from reference)
#define NL 4          // layers
#define NB 16         // batch of weight samples
#define NN 256        // base width
#define NC 16         // channels
#define JC 8          // j-chunks for row-sum partials

#define SW_OUT 16777216ull  // B*N*N*C  (ow_i elements)
#define SB_OUT 65536ull     // B*N*C    (ob_i elements)

typedef float v2f __attribute__((ext_vector_type(2)));
typedef float v8f __attribute__((ext_vector_type(8)));

// ---------------------------------------------------------------------------
// Kernel 1: streaming reduction over w_i.
// Block = (layer i, batch b, j-chunk of 32 rows), 256 threads.
// Produces: csum (mean over k) for each (b,j,c), and partial row-sums
// (sum over the chunk's 32 j's) for each (b,k,c) into scratch.
// ---------------------------------------------------------------------------
__global__ __launch_bounds__(256) void red_kernel(
    const float* __restrict__ w0, const float* __restrict__ w1,
    const float* __restrict__ w2, const float* __restrict__ w3,
    float* __restrict__ partial, float* __restrict__ csum) {
  int bid = blockIdx.x;
  int i  = bid >> 7;          // 16*8 = 128 blocks per layer
  int b  = (bid >> 3) & 15;
  int jc = bid & 7;
  const float* w = (i==0)?w0:(i==1)?w1:(i==2)?w2:w3;
  int t = threadIdx.x;
  int c = t & 15, kb = t >> 4;
  float racc[16];
#pragma unroll
  for (int m = 0; m < 16; ++m) racc[m] = 0.f;
  __shared__ float red[256];
  for (int jj = 0; jj < 32; ++jj) {
    int j = jc * 32 + jj;
    const float* row = w + (size_t)(b * 256 + j) * 256 * 16;
    float local = 0.f;
#pragma unroll
    for (int m = 0; m < 16; ++m) {
      // element (k = m*16+kb, c): flat = m*256 + t  (fully coalesced 1KB reads)
      float v = row[m * 256 + t];
      local += v;
      racc[m] += v;
    }
    red[t] = local;
    __syncthreads();
    if (t < 16) {
      float s = 0.f;
#pragma unroll
      for (int g = 0; g < 16; ++g) s += red[g * 16 + t];
      csum[(size_t)((i * 16 + b) * 256 + j) * 16 + t] = s * (1.0f / 256.0f);
    }
    __syncthreads();
  }
#pragma unroll
  for (int m = 0; m < 16; ++m) {
    int k = m * 16 + kb;
    partial[(size_t)(((i * 16 + b) * JC + jc) * 256 + k) * 16 + c] = racc[m];
  }
}

// ---------------------------------------------------------------------------
// Kernel 2: SIREN evaluated at the 100 coordinate points -> term matrices.
// Block = one point, 64 threads (one per hidden unit).
// terms layout: point p -> 256 floats [c*16+d], points ordered
//   [0,64): global (t,a,l); [64,88): diag (s,l); [88,94): p1; [94,100): m1
// ---------------------------------------------------------------------------
__global__ __launch_bounds__(64) void siren_kernel(
    const float* __restrict__ sw0, const float* __restrict__ sb0,
    const float* __restrict__ sw1, const float* __restrict__ sb1,
    const float* __restrict__ sw2, const float* __restrict__ sb2,
    const float* __restrict__ swo, const float* __restrict__ sbo,
    float* __restrict__ terms) {
  const float lay[4] = {-1.f, -1.f / 3.f, 1.f / 3.f, 1.f};
  int p = blockIdx.x;
  int t = threadIdx.x;
  float x0, x1;
  int term;
  if (p < 64) {
    int tt = p >> 4, a = (p >> 2) & 3, l = p & 3;
    x0 = lay[a]; x1 = lay[l]; term = tt;
  } else if (p < 88) {
    int q = p - 64; int s = q >> 2, l = q & 3;
    x0 = lay[l]; x1 = lay[l]; term = 4 + s;
  } else if (p < 94) {
    int q = p - 88; int s = q / 3, l = q % 3;
    x0 = lay[l]; x1 = lay[l + 1]; term = 10 + s;
  } else {
    int q = p - 94; int s = q / 3, l = q % 3;
    x0 = lay[l + 1]; x1 = lay[l]; term = 12 + s;
  }
  float x2 = -1.0f + 2.0f * (float)term / 13.0f;

  __shared__ float h[64];
  float a0 = x0 * sw0[0 * 64 + t] + x1 * sw0[1 * 64 + t] + x2 * sw0[2 * 64 + t] + sb0[t];
  h[t] = sinf(30.0f * a0);
  __syncthreads();
  float a1 = sb1[t];
  for (int u = 0; u < 64; ++u) a1 += h[u] * sw1[u * 64 + t];
  __syncthreads();
  h[t] = sinf(a1);
  __syncthreads();
  float a2 = sb2[t];
  for (int u = 0; u < 64; ++u) a2 += h[u] * sw2[u * 64 + t];
  __syncthreads();
  h[t] = sinf(a2);
  __syncthreads();
  const float inv_norm = 0.06454972243679028f;  // 1/sqrt((2*4+7)*16)
#pragma unroll
  for (int r = 0; r < 4; ++r) {
    int d = r * 64 + t;
    float o = sbo[d];
    for (int u = 0; u < 64; ++u) o += h[u] * swo[u * 256 + d];
    terms[(size_t)p * 256 + d] = o * inv_norm;
  }
}

// ---------------------------------------------------------------------------
// Kernel 3: finalize reductions: rsum means, allsums, sums_b.
// Block = (i,b), 256 threads.
// ---------------------------------------------------------------------------
__global__ __launch_bounds__(256) void fin1_kernel(
    const float* __restrict__ partial,
    const float* __restrict__ bs0, const float* __restrict__ bs1,
    const float* __restrict__ bs2, const float* __restrict__ bs3,
    float* __restrict__ rsum, float* __restrict__ allsums,
    float* __restrict__ sumsb) {
  int bid = blockIdx.x;
  int i = bid >> 4, b = bid & 15;
  const float* bp = (i==0)?bs0:(i==1)?bs1:(i==2)?bs2:bs3;
  int t = threadIdx.x;
  int c = t & 15, kb = t >> 4;
  __shared__ float red[256];
  float localsum = 0.f;
#pragma unroll
  for (int m = 0; m < 16; ++m) {
    int k = m * 16 + kb;
    float s = 0.f;
    for (int jc = 0; jc < JC; ++jc)
      s += partial[(size_t)(((i * 16 + b) * JC + jc) * 256 + k) * 16 + c];
    float mean = s * (1.0f / 256.0f);
    rsum[(size_t)((i * 16 + b) * 256 + k) * 16 + c] = mean;
    localsum += mean;
  }
  red[t] = localsum;
  __syncthreads();
  if (t < 16) {
    float s = 0.f;
#pragma unroll
    for (int g = 0; g < 16; ++g) s += red[g * 16 + t];
    allsums[(b * 16 + t) * 4 + i] = s * (1.0f / 256.0f);
  }
  __syncthreads();
  float lb = 0.f;
#pragma unroll
  for (int m = 0; m < 16; ++m) {
    int n = m * 16 + kb;
    lb += bp[(size_t)(b * 256 + n) * 16 + c];
  }
  red[t] = lb;
  __syncthreads();
  if (t < 16) {
    float s = 0.f;
#pragma unroll
    for (int g = 0; g < 16; ++g) s += red[g * 16 + t];
    sumsb[(b * 16 + t) * 4 + i] = s * (1.0f / 256.0f);
  }
}

// ---------------------------------------------------------------------------
// Kernel 4: build fused rowterm/colterm (+scalar) and write ob outputs.
// Block = (i,b), 256 threads.
// ---------------------------------------------------------------------------
__global__ __launch_bounds__(256) void fin2_kernel(
    const float* __restrict__ terms,
    const float* __restrict__ rsum, const float* __restrict__ csum,
    const float* __restrict__ allsums, const float* __restrict__ sumsb,
    const float* __restrict__ bs0, const float* __restrict__ bs1,
    const float* __restrict__ bs2, const float* __restrict__ bs3,
    float* __restrict__ rowterm, float* __restrict__ colterm,
    float* __restrict__ out) {
  int bid = blockIdx.x;
  int i = bid >> 4, b = bid & 15;
  const float* bcur  = (i==0)?bs0:(i==1)?bs1:(i==2)?bs2:bs3;
  const float* bprev = (i==1)?bs0:(i==2)?bs1:(i==3)?bs2:bs0;  // used only if i>0

  const float* G  = terms;                 // global: ((t*4+i)*4+l)*256 + c*16+d
  const float* D0 = terms + 64 * 256;      // diag:   (s*4+i)*256
  const float* P0 = terms + 88 * 256;      // p1:     (s*3+i)*256
  const float* M0 = terms + 94 * 256;      // m1:     (s*3+(i-1))*256
  const float* t_w_otherw = G + (size_t)((0 * 4 + i) * 4) * 256;
  const float* t_w_otherb = G + (size_t)((1 * 4 + i) * 4) * 256;
  const float* t_b_otherw = G + (size_t)((2 * 4 + i) * 4) * 256;
  const float* t_b_otherb = G + (size_t)((3 * 4 + i) * 4) * 256;
  const float* t_w_wrow = D0 + (size_t)(0 * 4 + i) * 256;
  const float* t_w_wcol = D0 + (size_t)(1 * 4 + i) * 256;
  const float* t_w_b    = D0 + (size_t)(3 * 4 + i) * 256;
  const float* t_b_wcol = D0 + (size_t)(4 * 4 + i) * 256;
  const float* t_bb     = D0 + (size_t)(5 * 4 + i) * 256;
  const float* t_w_wp1  = P0 + (size_t)(0 * 3 + i) * 256;
  const float* t_b_wp1  = P0 + (size_t)(1 * 3 + i) * 256;
  const float* t_w_wm1  = M0 + (size_t)(0 * 3 + (i - 1)) * 256;
  const float* t_w_bm1  = M0 + (size_t)(1 * 3 + (i - 1)) * 256;

  __shared__ float sc_w[16], sc_b[16];
  int t = threadIdx.x;
  if (t < 16) {
    int d = t;
    float sw = 0.f, sb = 0.f;
    for (int c = 0; c < 16; ++c)
      for (int l = 0; l < 4; ++l) {
        float as = allsums[(b * 16 + c) * 4 + l];
        float sv = sumsb[(b * 16 + c) * 4 + l];
        sw += as * t_w_otherw[l * 256 + c * 16 + d] + sv * t_w_otherb[l * 256 + c * 16 + d];
        sb += as * t_b_otherw[l * 256 + c * 16 + d] + sv * t_b_otherb[l * 256 + c * 16 + d];
      }
    sc_w[d] = sw; sc_b[d] = sb;
  }
  __syncthreads();

  int d = t & 15, nb = t >> 4;
  size_t ob_off = (size_t)i * (SW_OUT + SB_OUT) + SW_OUT;
#pragma unroll 1
  for (int m = 0; m < 16; ++m) {
    int n = m * 16 + nb;
    const float* rs = rsum + (size_t)((i * 16 + b) * 256 + n) * 16;
    const float* cs = csum + (size_t)((i * 16 + b) * 256 + n) * 16;
    const float* bv = bcur + (size_t)(b * 256 + n) * 16;
    // rowterm[b,k=n,d]
    float rt = 0.f;
    for (int c = 0; c < 16; ++c)
      rt += rs[c] * t_w_wrow[c * 16 + d] + bv[c] * t_w_b[c * 16 + d];
    if (i < 3) {
      const float* cs1 = csum + (size_t)(((i + 1) * 16 + b) * 256 + n) * 16;
      for (int c = 0; c < 16; ++c) rt += cs1[c] * t_w_wp1[c * 16 + d];
    }
    rowterm[(size_t)((i * 16 + b) * 256 + n) * 16 + d] = rt;
    // colterm[b,j=n,d] (scalar folded in)
    float ct = sc_w[d];
    for (int c = 0; c < 16; ++c) ct += cs[c] * t_w_wcol[c * 16 + d];
    if (i > 0) {
      const float* rsm = rsum + (size_t)(((i - 1) * 16 + b) * 256 + n) * 16;
      const float* bm  = bprev + (size_t)(b * 256 + n) * 16;
      for (int c = 0; c < 16; ++c)
        ct += rsm[c] * t_w_wm1[c * 16 + d] + bm[c] * t_w_bm1[c * 16 + d];
    }
    colterm[(size_t)((i * 16 + b) * 256 + n) * 16 + d] = ct;
    // ob[b,j=n,d] -> d_out
    float ob = sc_b[d];
    for (int c = 0; c < 16; ++c)
      ob += bv[c] * t_bb[c * 16 + d] + rs[c] * t_b_wcol[c * 16 + d];
    if (i < 3) {
      const float* cs1 = csum + (size_t)(((i + 1) * 16 + b) * 256 + n) * 16;
      for (int c = 0; c < 16; ++c) ob += cs1[c] * t_b_wp1[c * 16 + d];
    }
    out[ob_off + (size_t)(b * 256 + n) * 16 + d] = ob;
  }
}

// ---------------------------------------------------------------------------
// Kernel 5 (main, memory-bound, WMMA f32):
// ow[b,j,k,d] = sum_c w[b,j,k,c]*T[c,d] + rowterm[b,k,d] + colterm[b,j,d]
// Wave handles one (i,b,j) row; 16 tiles of 16 k's; each tile:
//   D(16x16) = X(16 elems x 16 c) * T + C, via 4 chained 16x16x4 f32 WMMAs.
// ---------------------------------------------------------------------------
__global__ __launch_bounds__(256) void main_kernel(
    const float* __restrict__ w0, const float* __restrict__ w1,
    const float* __restrict__ w2, const float* __restrict__ w3,
    const float* __restrict__ terms,
    const float* __restrict__ rowterm, const float* __restrict__ colterm,
    float* __restrict__ out) {
  int bid = blockIdx.x;
  int i = bid >> 9;                   // 512 blocks per layer
  int b = (bid >> 5) & 15;
  int j = ((bid & 31) << 3) + (threadIdx.x >> 5);   // 8 waves -> 8 j rows
  int lane = threadIdx.x & 31;
  int hl = lane >> 4;                 // half: selects K-pair (A/B layout)
  int ln = lane & 15;                 // M for A, N(=d) for B/C/D

  const float* w = (i==0)?w0:(i==1)?w1:(i==2)?w2:w3;
  const float* T = terms + 64 * 256 + (size_t)(2 * 4 + i) * 256;  // t_w_w[i]

  // B fragment: 4x16 chunk q covers rows c=4q..4q+3.
  // VGPR0: K=4q+2*hl, VGPR1: K=4q+2*hl+1, cols d=ln.
  v2f tb[4];
#pragma unroll
  for (int q = 0; q < 4; ++q) {
    tb[q].x = T[(4 * q + 2 * hl) * 16 + ln];
    tb[q].y = T[(4 * q + 2 * hl + 1) * 16 + ln];
  }
  float colscal = colterm[(size_t)((i * 16 + b) * 256 + j) * 16 + ln];
  const float* rt = rowterm + (size_t)((i * 16 + b) * 256) * 16;

  size_t base  = (size_t)(b * 256 + j) * 256 * 16;          // w / ow row base
  size_t obase = (size_t)i * (SW_OUT + SB_OUT) + base;

  for (int kt = 0; kt < 16; ++kt) {
    int k0 = kt * 16;
    // C fragment: VGPR r holds (M=r+8*hl, N=ln)
    v8f acc;
#pragma unroll
    for (int r = 0; r < 8; ++r)
      acc[r] = rt[(size_t)(k0 + r + 8 * hl) * 16 + ln] + colscal;
    // A fragment per chunk q: lane (hl,ln) holds M=ln, K=4q+2*hl, +1
#pragma unroll
    for (int q = 0; q < 4; ++q) {
      v2f a = *(const v2f*)(w + base + (size_t)(k0 + ln) * 16 + 4 * q + 2 * hl);
      acc = __builtin_amdgcn_wmma_f32_16x16x4_f32(
          false, a, false, tb[q], (short)0, acc, false, false);
    }
#pragma unroll
    for (int r = 0; r < 8; ++r)
      out[obase + (size_t)(k0 + r + 8 * hl) * 16 + ln] = acc[r];
  }
}

// ---------------------------------------------------------------------------
extern "C" void kernel_launch(void* const* d_in, const int* in_sizes, int n_in,
                              void* d_out, int out_size, void* d_ws, size_t ws_size,
                              hipStream_t stream) {
  // setup_inputs() dict order: w0,b0,w1,b1,w2,b2,w3,b3, s_w0,s_b0,s_w1,s_b1,s_w2,s_b2,s_wo,s_bo
  const float* w0 = (const float*)d_in[0];
  const float* b0 = (const float*)d_in[1];
  const float* w1 = (const float*)d_in[2];
  const float* b1 = (const float*)d_in[3];
  const float* w2 = (const float*)d_in[4];
  const float* b2 = (const float*)d_in[5];
  const float* w3 = (const float*)d_in[6];
  const float* b3 = (const float*)d_in[7];
  const float* sw0 = (const float*)d_in[8];
  const float* sb0 = (const float*)d_in[9];
  const float* sw1 = (const float*)d_in[10];
  const float* sb1 = (const float*)d_in[11];
  const float* sw2 = (const float*)d_in[12];
  const float* sb2 = (const float*)d_in[13];
  const float* swo = (const float*)d_in[14];
  const float* sbo = (const float*)d_in[15];
  float* outp = (float*)d_out;

  // Scratch layout (floats): ~12.7 MB total
  float* wsf     = (float*)d_ws;
  float* terms   = wsf;                       // 25600
  float* partial = terms + 25600;             // 4*16*JC*256*16 = 2097152
  float* rsum    = partial + (size_t)4*16*JC*256*16;  // 262144
  float* csum    = rsum + (size_t)4*16*256*16;        // 262144
  float* allsums = csum + (size_t)4*16*256*16;        // 1024
  float* sumsb   = allsums + 1024;                    // 1024
  float* rowt    = sumsb + 1024;                      // 262144
  float* colt    = rowt + (size_t)4*16*256*16;        // 262144

  red_kernel<<<512, 256, 0, stream>>>(w0, w1, w2, w3, partial, csum);
  siren_kernel<<<100, 64, 0, stream>>>(sw0, sb0, sw1, sb1, sw2, sb2, swo, sbo, terms);
  fin1_kernel<<<64, 256, 0, stream>>>(partial, b0, b1, b2, b3, rsum, allsums, sumsb);
  fin2_kernel<<<64, 256, 0, stream>>>(terms, rsum, csum, allsums, sumsb,
                                      b0, b1, b2, b3, rowt, colt, outp);
  main_kernel<<<2048, 256, 0, stream>>>(w0, w1, w2, w3, terms, rowt, colt, outp);
}